// Mahalanobis_mask_61993557950769
// MI455X (gfx1250) — compile-verified
//
#include <hip/hip_runtime.h>
#include <math.h>

// Mahalanobis mask, fused:  B=8, X=256, C=256
//   s = wa*amp + wp*ph            (B,X,C)
//   G[b] = S_b^T S_b via v_wmma_f32_16x16x4_f32
//   dist[b,c,d] = diagA[d]^2 * (n[c]+n[d]-2G)
//   p, logit, 2-class gumbel hard sample, mean over b -> mask (C,C)

typedef __attribute__((ext_vector_type(2))) float v2f;
typedef __attribute__((ext_vector_type(8))) float v8f;

#define CC 256
#define XX 256
#define NB 8
#define CHUNK 32
#define LPAD 264   // padded LDS row stride (floats); keeps half-wave frag reads on disjoint banks

__global__ void zero_out_kernel(float* __restrict__ out) {
    out[blockIdx.x * blockDim.x + threadIdx.x] = 0.0f;
}

__global__ __launch_bounds__(512, 1)
void mahalanobis_mask_kernel(const float* __restrict__ amp,
                             const float* __restrict__ ph,
                             const float* __restrict__ A,
                             const float* __restrict__ w_amp,
                             const float* __restrict__ w_ph,
                             const float* __restrict__ gu,
                             float* __restrict__ out) {
    __shared__ float s_lds[CHUNK * LPAD];   // staged s chunk (32 x 256, padded)
    __shared__ float normAcc[CC];           // column sum of squares
    __shared__ float diagA2[CC];            // diag(A)^2
    __shared__ float rowMinWS[16 * 16];     // per-wave row mins
    __shared__ float distMin[16];           // strip row mins (over all d != c)

    const int t    = threadIdx.x;
    const int w    = t >> 5;     // wave id 0..15 -> d tile
    const int lane = t & 31;
    const int half = lane >> 4;  // 0: K pair {0,1} / rows 0..7 ; 1: K pair {2,3} / rows 8..15
    const int mn   = lane & 15;
    const int c0   = blockIdx.x * 16;
    const int bb   = blockIdx.y;
    const int d0   = w * 16;

    const float wa = w_amp[0];
    const float wp = w_ph[0];

    if (t < CC) {
        normAcc[t] = 0.0f;
        float ad = A[t * (CC + 1)];
        diagA2[t] = ad * ad;
    }

    const float* ampB = amp + (size_t)bb * XX * CC;
    const float* phB  = ph  + (size_t)bb * XX * CC;

    v8f acc = {0.f, 0.f, 0.f, 0.f, 0.f, 0.f, 0.f, 0.f};

    for (int x0 = 0; x0 < XX; x0 += CHUNK) {
        __syncthreads();  // previous chunk fully consumed
        // ---- stage s = wa*amp + wp*ph for CHUNK rows (float4 per op, coalesced) ----
        #pragma unroll
        for (int j = 0; j < (CHUNK * CC) / (512 * 4); ++j) {
            int i  = (t + 512 * j) * 4;
            int xr = i >> 8;     // row within chunk
            int cc = i & 255;
            const float4 a4 = *(const float4*)(ampB + (size_t)(x0 + xr) * CC + cc);
            const float4 p4 = *(const float4*)(phB  + (size_t)(x0 + xr) * CC + cc);
            float4 s4;
            s4.x = wa * a4.x + wp * p4.x;
            s4.y = wa * a4.y + wp * p4.y;
            s4.z = wa * a4.z + wp * p4.z;
            s4.w = wa * a4.w + wp * p4.w;
            *(float4*)(&s_lds[xr * LPAD + cc]) = s4;
        }
        __syncthreads();
        // ---- accumulate column norms for this chunk (one owner thread per column) ----
        if (t < CC) {
            float an = 0.0f;
            #pragma unroll 8
            for (int xr = 0; xr < CHUNK; ++xr) {
                float v = s_lds[xr * LPAD + t];
                an += v * v;
            }
            normAcc[t] += an;
        }
        // ---- WMMA: 8 k-steps of K=4 over this chunk ----
        #pragma unroll
        for (int kk = 0; kk < CHUNK / 4; ++kk) {
            int kr = kk * 4 + 2 * half;
            v2f a, b;
            a.x = s_lds[kr * LPAD + c0 + mn];
            a.y = s_lds[(kr + 1) * LPAD + c0 + mn];
            b.x = s_lds[kr * LPAD + d0 + mn];
            b.y = s_lds[(kr + 1) * LPAD + d0 + mn];
            acc = __builtin_amdgcn_wmma_f32_16x16x4_f32(
                /*neg_a=*/false, a, /*neg_b=*/false, b,
                /*c_mod=*/(short)0, acc, /*reuse_a=*/false, /*reuse_b=*/false);
        }
    }
    __syncthreads();

    // ---- dist + per-row min (diag excluded) ----
    float distv[8];
    #pragma unroll
    for (int r = 0; r < 8; ++r) {
        int m = r + 8 * half;
        int c = c0 + m;
        int d = d0 + mn;
        float g    = acc[r];
        float dist = diagA2[d] * (normAcc[c] + normAcc[d] - 2.0f * g);
        distv[r]   = dist;
        float dm = (d == c) ? 3.402823466e38f : dist;
        #pragma unroll
        for (int s = 1; s < 16; s <<= 1)
            dm = fminf(dm, __shfl_xor(dm, s, 32));  // stays within 16-lane half
        if (mn == 0) rowMinWS[w * 16 + m] = dm;
    }
    __syncthreads();
    if (t < 16) {
        float mv = rowMinWS[t];
        #pragma unroll
        for (int ww = 1; ww < 16; ++ww) mv = fminf(mv, rowMinWS[ww * 16 + t]);
        distMin[t] = mv;
    }
    __syncthreads();

    // ---- epilogue: p -> logit -> gumbel argmax -> mean over b ----
    const float2* gu2 = (const float2*)gu;
    #pragma unroll
    for (int r = 0; r < 8; ++r) {
        int m = r + 8 * half;
        int c = c0 + m;
        int d = d0 + mn;
        // exp_dist/exp_max == (minDist + eps)/(dist + eps); diag forced to 0.99
        float p = (d == c) ? 0.99f
                           : 0.99f * (distMin[m] + 1e-10f) / (distv[r] + 1e-10f);
        float logit = __logf(p / (1.0f - p));
        float2 u = gu2[((size_t)bb * CC + c) * CC + d];
        float g0 = -__logf(-__logf(u.x));
        float g1 = -__logf(-__logf(u.y));
        // straight-through sample is numerically hard0 = (logit+g0 > -logit+g1)
        float hard = (2.0f * logit + g0 - g1 > 0.0f) ? 1.0f : 0.0f;
        atomicAdd(&out[c * CC + d], hard * 0.125f);
    }
}

extern "C" void kernel_launch(void* const* d_in, const int* in_sizes, int n_in,
                              void* d_out, int out_size, void* d_ws, size_t ws_size,
                              hipStream_t stream) {
    const float* amp  = (const float*)d_in[0];
    const float* ph   = (const float*)d_in[1];
    const float* A    = (const float*)d_in[2];
    const float* wamp = (const float*)d_in[3];
    const float* wph  = (const float*)d_in[4];
    const float* gu   = (const float*)d_in[5];
    float* out = (float*)d_out;

    zero_out_kernel<<<dim3(64), dim3(1024), 0, stream>>>(out);  // 65536 floats
    mahalanobis_mask_kernel<<<dim3(CC / 16, NB), dim3(512), 0, stream>>>(
        amp, ph, A, wamp, wph, gu, out);
}